// TtSqueezeExcitation_79027398246939
// MI455X (gfx1250) — compile-verified
//
#include <hip/hip_runtime.h>

typedef float v2f __attribute__((ext_vector_type(2)));
typedef float v8f __attribute__((ext_vector_type(8)));
typedef float f4  __attribute__((ext_vector_type(4)));

#define B_    16
#define C_    384
#define CR_   96
#define HW_   12544      // 112*112
#define HW4_  3136       // HW_/4
#define BC_   (B_ * C_)  // 6144

// Padded LDS pitches: even (8B-aligned float2 rows) and not a multiple of 64
// (bank-conflict-free across the 16 lanes reading different M rows).
#define MPITCH 386
#define TPITCH 98

// ---------------------------------------------------------------------------
// Kernel 1: per-(b,c) plane sum.  One block = one 112x112 plane (3136 float4).
// ---------------------------------------------------------------------------
__global__ __launch_bounds__(256) void se_pool_kernel(const float* __restrict__ x,
                                                      float* __restrict__ sums) {
    const int bc  = blockIdx.x;
    const int tid = threadIdx.x;
    const f4* xp = (const f4*)(x + (size_t)bc * HW_);

    float a = 0.0f;
    for (int i = tid; i < HW4_; i += 256) {
        f4 v = xp[i];
        a += (v.x + v.y) + (v.z + v.w);
    }
    // wave32 reduction
#pragma unroll
    for (int off = 16; off > 0; off >>= 1)
        a += __shfl_xor(a, off, 32);

    __shared__ float part[8];
    const int wave = tid >> 5;
    const int lane = tid & 31;
    if (lane == 0) part[wave] = a;
    __syncthreads();
    if (tid == 0) {
        float t = 0.0f;
#pragma unroll
        for (int w = 0; w < 8; ++w) t += part[w];
        sums[bc] = t;
    }
}

// ---------------------------------------------------------------------------
// Kernel 2: mean -> fc1(+ReLU) -> fc2(+bias) -> hardsigmoid, via
// V_WMMA_F32_16X16X4_F32.  Single block, 256 threads = 8 waves.
//
// A-fragment (16x4 f32): lane L holds M = L&15; VGPR0/1 = K = koff, koff+1
// where koff = (L>=16) ? 2 : 0.  B-fragment (4x16) mirrors: lane holds N=L&15,
// K = koff, koff+1.  D (16x16 f32): acc[r] is M = r + 8*(L>=16), N = L&15.
// ---------------------------------------------------------------------------
__global__ __launch_bounds__(256) void se_fc_kernel(const float* __restrict__ sums,
                                                    const float* __restrict__ w1,
                                                    const float* __restrict__ b1,
                                                    const float* __restrict__ w2,
                                                    const float* __restrict__ b2,
                                                    float* __restrict__ scale) {
    __shared__ float mean[B_ * MPITCH];
    __shared__ float t1[B_ * TPITCH];

    const int tid  = threadIdx.x;
    const int wave = tid >> 5;
    const int lane = tid & 31;
    const int nl   = lane & 15;
    const int hi   = lane >> 4;   // 0 or 1
    const int koff = hi * 2;

    // stage 0: means into LDS
    const float inv_hw = 1.0f / (float)HW_;
    for (int i = tid; i < BC_; i += 256) {
        const int b = i / C_;
        const int c = i - b * C_;
        mean[b * MPITCH + c] = sums[i] * inv_hw;
    }
    __syncthreads();

    // stage 1: fc1  [16x384] @ B[384x96] where B[c][r] = w1[r*384 + c]
    if (wave < 6) {
        const int ncol = wave * 16 + nl;   // r index, 0..95
        v8f acc = {};
        for (int k0 = 0; k0 < C_; k0 += 4) {
            const float* ap = &mean[nl * MPITCH + k0 + koff];
            const float* bp = &w1[(size_t)ncol * C_ + k0 + koff];
            v2f a, b;
            a.x = ap[0]; a.y = ap[1];
            b.x = bp[0]; b.y = bp[1];
            acc = __builtin_amdgcn_wmma_f32_16x16x4_f32(
                false, a, false, b, (short)0, acc, false, false);
        }
        const float bias = b1[ncol];
#pragma unroll
        for (int r = 0; r < 8; ++r) {
            const int mrow = r + hi * 8;
            const float v = acc[r] + bias;
            t1[mrow * TPITCH + ncol] = v > 0.0f ? v : 0.0f;
        }
    }
    __syncthreads();

    // stage 2: fc2  [16x96] @ B[96x384] where B[r][c] = w2[c*96 + r]
#pragma unroll
    for (int j = 0; j < 3; ++j) {
        const int tile = wave + j * 8;     // 0..23
        const int ncol = tile * 16 + nl;   // c index, 0..383
        v8f acc = {};
        for (int k0 = 0; k0 < CR_; k0 += 4) {
            const float* ap = &t1[nl * TPITCH + k0 + koff];
            const float* bp = &w2[(size_t)ncol * CR_ + k0 + koff];
            v2f a, b;
            a.x = ap[0]; a.y = ap[1];
            b.x = bp[0]; b.y = bp[1];
            acc = __builtin_amdgcn_wmma_f32_16x16x4_f32(
                false, a, false, b, (short)0, acc, false, false);
        }
        const float bias = b2[ncol];
#pragma unroll
        for (int r = 0; r < 8; ++r) {
            const int mrow = r + hi * 8;
            float v = acc[r] + bias;
            v = v * (1.0f / 6.0f) + 0.5f;          // hardsigmoid
            v = fminf(fmaxf(v, 0.0f), 1.0f);
            scale[mrow * C_ + ncol] = v;
        }
    }
}

// ---------------------------------------------------------------------------
// Kernel 3: out = x * scale[b,c].  Reverse plane order to hit the tail of x
// still resident in the 192MB L2 from kernel 1.  Non-temporal stores keep the
// streaming output from evicting those x lines.
// ---------------------------------------------------------------------------
__global__ __launch_bounds__(256) void se_apply_kernel(const float* __restrict__ x,
                                                       const float* __restrict__ scale,
                                                       float* __restrict__ out) {
    const int bc = (BC_ - 1) - (int)blockIdx.y;
    const int i  = ((int)gridDim.x - 1 - (int)blockIdx.x) * 256 + (int)threadIdx.x;
    if (i >= HW4_) return;

    const float s = scale[bc];
    const f4* xp = (const f4*)(x + (size_t)bc * HW_);
    f4*       op = (f4*)(out + (size_t)bc * HW_);

    f4 v = xp[i];
    f4 r;
    r.x = v.x * s;
    r.y = v.y * s;
    r.z = v.z * s;
    r.w = v.w * s;
    __builtin_nontemporal_store(r, op + i);
}

// ---------------------------------------------------------------------------
extern "C" void kernel_launch(void* const* d_in, const int* in_sizes, int n_in,
                              void* d_out, int out_size, void* d_ws, size_t ws_size,
                              hipStream_t stream) {
    (void)in_sizes; (void)n_in; (void)out_size; (void)ws_size;

    const float* x  = (const float*)d_in[0];   // [16,384,112,112]
    const float* w1 = (const float*)d_in[1];   // [96,384,1,1]
    const float* b1 = (const float*)d_in[2];   // [96]
    const float* w2 = (const float*)d_in[3];   // [384,96,1,1]
    const float* b2 = (const float*)d_in[4];   // [384]
    float*       out = (float*)d_out;

    float* sums  = (float*)d_ws;        // 6144 floats
    float* scale = sums + BC_;          // 6144 floats

    se_pool_kernel<<<BC_, 256, 0, stream>>>(x, sums);
    se_fc_kernel<<<1, 256, 0, stream>>>(sums, w1, b1, w2, b2, scale);

    dim3 g3((HW4_ + 255) / 256, BC_);   // 13 x 6144
    se_apply_kernel<<<g3, 256, 0, stream>>>(x, scale, out);
}